// GAT_36867999269229
// MI455X (gfx1250) — compile-verified
//
#include <hip/hip_runtime.h>
#include <math.h>

#define NN 50000
#define NE 800000
#define NF 256
#define ND 32
#define NH 4
#define LRELU 0.2f

typedef __attribute__((ext_vector_type(2))) float v2f;
typedef __attribute__((ext_vector_type(8))) float v8f;

// ---------- monotonic float <-> uint encoding for atomicMax ----------
__device__ __forceinline__ unsigned fenc(float f) {
    unsigned u = __float_as_uint(f);
    return (u & 0x80000000u) ? ~u : (u | 0x80000000u);
}
__device__ __forceinline__ float fdec(unsigned u) {
    unsigned v = (u & 0x80000000u) ? (u & 0x7FFFFFFFu) : ~u;
    return __uint_as_float(v);
}

// ---------------------------------------------------------------------
// WMMA f32 projection: C[M x ncols] = A[M x K] @ B
// B is stored as per-head blocks: element (k, c) with head=c/cph, j=c%cph
// lives at B[head*K*cph + k*cph + j].  One wave computes a 16x16 tile via
// V_WMMA_F32_16X16X4_F32 accumulation (exact f32 math).
// Requires M % 16 == 0, ncols % 16 == 0, K % 4 == 0, cph % 16 == 0.
// ---------------------------------------------------------------------
__global__ void gat_wmma_proj(const float* __restrict__ A,
                              const float* __restrict__ B,
                              float* __restrict__ C,
                              int K, int ncols, int cph)
{
    const int lane   = threadIdx.x;                       // 0..31
    const int tile_n = blockIdx.y * blockDim.y + threadIdx.y;
    const int m0     = blockIdx.x * 16;
    const int col0   = tile_n * 16;

    const int head = col0 / cph;
    const int j0   = col0 % cph;
    const float* Bp = B + (size_t)head * K * cph + j0;

    const int mrow = m0 + (lane & 15);        // A row for this lane
    const int kofs = 2 * (lane >> 4);         // K sub-offset per A/B layout
    const int jl   = lane & 15;               // B local column

    v8f acc = {};
    for (int k0 = 0; k0 < K; k0 += 4) {
        // A fragment: lane<16 -> (K=k0,k0+1); lane>=16 -> (K=k0+2,k0+3)
        v2f a = *(const v2f*)(A + (size_t)mrow * K + (k0 + kofs));
        // B fragment mirrors A: rows k0+kofs, k0+kofs+1 at column col0+jl
        int kb = k0 + kofs;
        v2f b;
        b.x = Bp[(size_t)kb * cph + jl];
        b.y = Bp[(size_t)(kb + 1) * cph + jl];
        acc = __builtin_amdgcn_wmma_f32_16x16x4_f32(
            /*neg_a=*/false, a, /*neg_b=*/false, b,
            /*c_mod=*/(short)0, acc, /*reuse_a=*/false, /*reuse_b=*/false);
    }

    // D layout: VGPR r holds M = r (lanes 0-15) / r+8 (lanes 16-31), N = lane&15
    const int crow = m0 + 8 * (lane >> 4);
    const int ccol = col0 + (lane & 15);
    #pragma unroll
    for (int r = 0; r < 8; ++r)
        C[(size_t)(crow + r) * ncols + ccol] = acc[r];
}

// ---------------------------------------------------------------------
// Per-node attention scalars: as[h][n] = h[n]·a[:d], ad[h][n] = h[n]·a[d:]
// ---------------------------------------------------------------------
__global__ void gat_alpha(const float* __restrict__ H,
                          const float* __restrict__ Avec,
                          float* __restrict__ asn,
                          float* __restrict__ adn,
                          int nheads, int d)
{
    int n = blockIdx.x * blockDim.x + threadIdx.x;
    if (n >= NN) return;
    const int ldh = nheads * d;
    for (int h = 0; h < nheads; ++h) {
        const float* hp = H + (size_t)n * ldh + h * d;
        const float* av = Avec + h * 2 * d;
        float s = 0.f, t = 0.f;
        for (int j = 0; j < d; ++j) {
            float v = hp[j];
            s += v * av[j];
            t += v * av[d + j];
        }
        asn[(size_t)h * NN + n] = s;
        adn[(size_t)h * NN + n] = t;
    }
}

__device__ __forceinline__ float edge_logit(const float* asn, const float* adn,
                                            int h, int s, int dn)
{
    float l = asn[(size_t)h * NN + s] + adn[(size_t)h * NN + dn];
    return l > 0.f ? l : LRELU * l;
}

// Pass A: segment max (keyed on src) via encoded uint atomicMax
__global__ void gat_edge_max(const int* __restrict__ src, const int* __restrict__ dst,
                             const float* __restrict__ asn, const float* __restrict__ adn,
                             unsigned* __restrict__ mx, int nheads)
{
    int tid = blockIdx.x * blockDim.x + threadIdx.x;
    if (tid >= NE * nheads) return;
    int e = tid % NE, h = tid / NE;
    int s = src[e];
    float l = edge_logit(asn, adn, h, s, dst[e]);
    atomicMax(&mx[(size_t)h * NN + s], fenc(l));
}

// Pass B: denom = segment_sum(exp(e - m))
__global__ void gat_edge_denom(const int* __restrict__ src, const int* __restrict__ dst,
                               const float* __restrict__ asn, const float* __restrict__ adn,
                               const unsigned* __restrict__ mx, float* __restrict__ dn,
                               int nheads)
{
    int tid = blockIdx.x * blockDim.x + threadIdx.x;
    if (tid >= NE * nheads) return;
    int e = tid % NE, h = tid / NE;
    int s = src[e];
    float l = edge_logit(asn, adn, h, s, dst[e]);
    float ex = __expf(l - fdec(mx[(size_t)h * NN + s]));
    atomicAdd(&dn[(size_t)h * NN + s], ex);
}

// Pass C: h_prime[src] += att * h[dst]
__global__ void gat_edge_agg(const int* __restrict__ src, const int* __restrict__ dst,
                             const float* __restrict__ asn, const float* __restrict__ adn,
                             const unsigned* __restrict__ mx, const float* __restrict__ dn,
                             const float* __restrict__ H, float* __restrict__ HP,
                             int nheads, int d)
{
    int tid = blockIdx.x * blockDim.x + threadIdx.x;
    if (tid >= NE * nheads) return;
    int e = tid % NE, h = tid / NE;
    int s = src[e], dnode = dst[e];
    float l = edge_logit(asn, adn, h, s, dnode);
    float att = __expf(l - fdec(mx[(size_t)h * NN + s])) / dn[(size_t)h * NN + s];

    const int ldh = nheads * d;
    const float* hd = H  + (size_t)dnode * ldh + h * d;
    float*       hp = HP + (size_t)s     * ldh + h * d;
    for (int j = 0; j < d; j += 4) {
        float4 v = *(const float4*)(hd + j);
        atomicAdd(hp + j + 0, att * v.x);
        atomicAdd(hp + j + 1, att * v.y);
        atomicAdd(hp + j + 2, att * v.z);
        atomicAdd(hp + j + 3, att * v.w);
    }
}

__global__ void gat_elu_inplace(float* __restrict__ x, int n)
{
    int i = blockIdx.x * blockDim.x + threadIdx.x;
    if (i >= n) return;
    float v = x[i];
    x[i] = v > 0.f ? v : (__expf(v) - 1.f);
}

__global__ void gat_elu_copy(const float* __restrict__ in, float* __restrict__ out, int n)
{
    int i = blockIdx.x * blockDim.x + threadIdx.x;
    if (i >= n) return;
    float v = in[i];
    out[i] = v > 0.f ? v : (__expf(v) - 1.f);
}

extern "C" void kernel_launch(void* const* d_in, const int* in_sizes, int n_in,
                              void* d_out, int out_size, void* d_ws, size_t ws_size,
                              hipStream_t stream)
{
    const float* x   = (const float*)d_in[0];   // [N, 256]
    const int*   src = (const int*)d_in[1];     // [E]
    const int*   dst = (const int*)d_in[2];     // [E]
    const float* Ws  = (const float*)d_in[3];   // [4, 256, 32]
    const float* As  = (const float*)d_in[4];   // [4, 64]
    const float* Wo  = (const float*)d_in[5];   // [128, 32]
    const float* Ao  = (const float*)d_in[6];   // [64]
    float* out = (float*)d_out;                 // [N, 32]

    // ---- workspace carve-up (256B aligned) ----
    char* w = (char*)d_ws;
    size_t off = 0;
    auto carve = [&](size_t bytes) {
        char* p = w + off;
        off += (bytes + 255) & ~(size_t)255;
        return p;
    };
    float*    H1  = (float*)carve((size_t)NN * 128 * 4);   // x @ Ws (concat heads)
    float*    HP1 = (float*)carve((size_t)NN * 128 * 4);   // layer-1 aggregation
    float*    AS1 = (float*)carve((size_t)NH * NN * 4);
    float*    AD1 = (float*)carve((size_t)NH * NN * 4);
    unsigned* MX1 = (unsigned*)carve((size_t)NH * NN * 4);
    float*    DN1 = (float*)carve((size_t)NH * NN * 4);
    float*    H2  = (float*)carve((size_t)NN * 32 * 4);    // elu(HP1) @ Wo
    float*    HP2 = (float*)carve((size_t)NN * 32 * 4);    // layer-2 aggregation
    float*    AS2 = (float*)carve((size_t)NN * 4);
    float*    AD2 = (float*)carve((size_t)NN * 4);
    unsigned* MX2 = (unsigned*)carve((size_t)NN * 4);
    float*    DN2 = (float*)carve((size_t)NN * 4);
    (void)ws_size; (void)in_sizes; (void)n_in; (void)out_size;

    // zero accumulators (graph-capture-safe async memsets)
    hipMemsetAsync(HP1, 0, (size_t)NN * 128 * 4, stream);
    hipMemsetAsync(MX1, 0, (size_t)NH * NN * 4, stream);
    hipMemsetAsync(DN1, 0, (size_t)NH * NN * 4, stream);
    hipMemsetAsync(HP2, 0, (size_t)NN * 32 * 4, stream);
    hipMemsetAsync(MX2, 0, (size_t)NN * 4, stream);
    hipMemsetAsync(DN2, 0, (size_t)NN * 4, stream);

    const int MT = NN / 16;                  // 3125 row tiles (exact)
    const int TPB = 256;
    const int gN  = (NN + TPB - 1) / TPB;
    const int gE4 = (NE * NH + TPB - 1) / TPB;
    const int gE1 = (NE + TPB - 1) / TPB;

    // ===== Layer 1 =====
    // H1 = x @ Ws  (M=50000, K=256, ncols=128, per-head col blocks of 32)
    gat_wmma_proj<<<dim3(MT, 2), dim3(32, 4), 0, stream>>>(x, Ws, H1, NF, 128, ND);
    gat_alpha<<<gN, TPB, 0, stream>>>(H1, As, AS1, AD1, NH, ND);
    gat_edge_max<<<gE4, TPB, 0, stream>>>(src, dst, AS1, AD1, MX1, NH);
    gat_edge_denom<<<gE4, TPB, 0, stream>>>(src, dst, AS1, AD1, MX1, DN1, NH);
    gat_edge_agg<<<gE4, TPB, 0, stream>>>(src, dst, AS1, AD1, MX1, DN1, H1, HP1, NH, ND);
    gat_elu_inplace<<<(NN * 128 + TPB - 1) / TPB, TPB, 0, stream>>>(HP1, NN * 128);

    // ===== Layer 2 =====
    // H2 = elu(HP1) @ Wo  (M=50000, K=128, ncols=32)
    gat_wmma_proj<<<dim3(MT, 1), dim3(32, 2), 0, stream>>>(HP1, Wo, H2, 128, 32, ND);
    gat_alpha<<<gN, TPB, 0, stream>>>(H2, Ao, AS2, AD2, 1, ND);
    gat_edge_max<<<gE1, TPB, 0, stream>>>(src, dst, AS2, AD2, MX2, 1);
    gat_edge_denom<<<gE1, TPB, 0, stream>>>(src, dst, AS2, AD2, MX2, DN2, 1);
    gat_edge_agg<<<gE1, TPB, 0, stream>>>(src, dst, AS2, AD2, MX2, DN2, H2, HP2, 1, ND);
    gat_elu_copy<<<(NN * 32 + TPB - 1) / TPB, TPB, 0, stream>>>(HP2, out, NN * 32);
}